// MemoNet_base_24051816857752
// MI455X (gfx1250) — compile-verified
//
#include <hip/hip_runtime.h>
#include <stdint.h>

// ---------------------------------------------------------------------------
// MemoNet get_memory_index for MI455X / gfx1250 (wave32, WMMA f32 16x16x4)
//   out = ( argsort(-cos_sim, axis=-1) [512x100000] int32,
//           cos_sim                    [512x100000] f32 )
// ---------------------------------------------------------------------------

typedef __attribute__((ext_vector_type(2))) float v2f;
typedef __attribute__((ext_vector_type(8))) float v8f;

#define BATCH 512
#define MEMN  100000
#define DIM   128

#define RADIX        256
#define SORT_THREADS 256
#define HIST_STRIDE  257   // pad to break LDS bank conflicts

// --------------------------------------------------------------------------
// Reciprocal L2 norms: one wave32 per row, float4 per lane, shuffle reduce.
// --------------------------------------------------------------------------
__global__ void memonet_inv_norm_kernel(const float* __restrict__ x,
                                        float* __restrict__ inv, int rows)
{
    const int wave = (int)((blockIdx.x * blockDim.x + threadIdx.x) >> 5);
    const int lane = (int)(threadIdx.x & 31);
    if (wave >= rows) return;
    const float4 v = *(const float4*)(x + (size_t)wave * DIM + lane * 4);
    float s = v.x * v.x + v.y * v.y + v.z * v.z + v.w * v.w;
    #pragma unroll
    for (int off = 16; off > 0; off >>= 1)
        s += __shfl_xor(s, off, 32);
    if (lane == 0)
        inv[wave] = 1.0f / fmaxf(sqrtf(s), 1e-12f);
}

// --------------------------------------------------------------------------
// Cosine similarity GEMM: C[512,100000] = S @ M^T, scaled by inv norms in the
// epilogue. One wave per 16x16 tile, 32x V_WMMA_F32_16X16X4_F32 over K=128.
// Epilogue also emits {descending-order sort key, column index} pairs.
// --------------------------------------------------------------------------
__global__ void memonet_cosine_wmma_kernel(const float* __restrict__ S,
                                           const float* __restrict__ M,
                                           const float* __restrict__ inv_s,
                                           const float* __restrict__ inv_m,
                                           float* __restrict__ out_w,
                                           uint2* __restrict__ keys)
{
    const int lane = (int)(threadIdx.x & 31);
    const int half = lane >> 4;     // K sub-pair selector per ISA layout
    const int lm   = lane & 15;     // M index (A) / N index (B)
    const int n0   = (int)blockIdx.x * 16;
    const int m0   = (int)blockIdx.y * 16;

    const float* aRow = S + (size_t)(m0 + lm) * DIM + 2 * half;
    const float* bRow = M + (size_t)(n0 + lm) * DIM + 2 * half;

    __builtin_prefetch(aRow, 0, 3);   // global_prefetch_b8
    __builtin_prefetch(bRow, 0, 3);

    v8f c = {};
    #pragma unroll
    for (int k = 0; k < DIM; k += 4) {
        v2f a = *(const v2f*)(aRow + k);   // A 16x4 f32: lane=M, half selects K pair
        v2f b = *(const v2f*)(bRow + k);   // B 4x16 f32: lane=N, half selects K pair
        c = __builtin_amdgcn_wmma_f32_16x16x4_f32(
                /*neg_a=*/false, a, /*neg_b=*/false, b,
                /*c_mod=*/(short)0, c, /*reuse_a=*/false, /*reuse_b=*/false);
    }

    // C/D layout: VGPR r, lanes 0-15 -> M=r, N=lane; lanes 16-31 -> M=8+r.
    const int   n    = n0 + lm;
    const float invN = inv_m[n];
    #pragma unroll
    for (int r = 0; r < 8; ++r) {
        const int m = m0 + r + 8 * half;
        const float w = c[r] * inv_s[m] * invN;
        const size_t o = (size_t)m * MEMN + n;
        out_w[o] = w;
        // order-preserving float->uint (ascending), complemented for descending
        unsigned u   = __float_as_uint(w);
        unsigned asc = ((int)u < 0) ? ~u : (u | 0x80000000u);
        keys[o] = make_uint2(~asc, (unsigned)n);
    }
}

// --------------------------------------------------------------------------
// Stable LSD radix pass: one workgroup per row, 256 threads, per-thread
// histograms in 264 KB of LDS (CDNA5 WGP has 320 KB). Atomic-free, stable.
// Final pass writes only the sorted index to d_out.
// --------------------------------------------------------------------------
__global__ void memonet_radix_pass_kernel(const uint2* __restrict__ src,
                                          uint2* __restrict__ dst,
                                          int* __restrict__ out_idx, // non-null => final
                                          int shift)
{
    extern __shared__ unsigned lds[];              // [RADIX][HIST_STRIDE] + [RADIX]
    unsigned* hist      = lds;
    unsigned* digitBase = lds + RADIX * HIST_STRIDE;

    const int    row     = (int)blockIdx.x;
    const int    t       = (int)threadIdx.x;
    const size_t rowBase = (size_t)row * MEMN;
    const int    chunk   = (MEMN + SORT_THREADS - 1) / SORT_THREADS;  // 391
    const int    p0      = t * chunk;
    const int    p1      = (p0 + chunk < MEMN) ? (p0 + chunk) : MEMN;

    for (int i = 0; i < HIST_STRIDE; ++i)          // thread t zeroes digit-row t
        hist[t * HIST_STRIDE + i] = 0;
    __syncthreads();

    // 1) count: column t of each digit row is exclusive to thread t (no atomics)
    for (int p = p0; p < p1; ++p) {
        const unsigned key = src[rowBase + p].x;
        const unsigned d   = (key >> shift) & (RADIX - 1);
        hist[d * HIST_STRIDE + t] += 1u;
    }
    __syncthreads();

    // 2) per-digit exclusive scan across threads (thread d owns digit d)
    {
        unsigned run = 0;
        for (int t2 = 0; t2 < SORT_THREADS; ++t2) {
            const unsigned v = hist[t * HIST_STRIDE + t2];
            hist[t * HIST_STRIDE + t2] = run;
            run += v;
        }
        digitBase[t] = run;                        // digit total
    }
    __syncthreads();

    // 3) exclusive scan over the 256 digit totals
    if (t == 0) {
        unsigned run = 0;
        for (int d = 0; d < RADIX; ++d) {
            const unsigned v = digitBase[d];
            digitBase[d] = run;
            run += v;
        }
    }
    __syncthreads();

    // 4) stable scatter (chunk order preserved -> overall stability)
    for (int p = p0; p < p1; ++p) {
        const uint2    kv     = src[rowBase + p];
        const unsigned d      = (kv.x >> shift) & (RADIX - 1);
        const unsigned within = hist[d * HIST_STRIDE + t];
        hist[d * HIST_STRIDE + t] = within + 1u;
        const unsigned pos = digitBase[d] + within;
        if (out_idx)
            out_idx[rowBase + pos] = (int)kv.y;
        else
            dst[rowBase + pos] = kv;
    }
}

// --------------------------------------------------------------------------
extern "C" void kernel_launch(void* const* d_in, const int* in_sizes, int n_in,
                              void* d_out, int out_size, void* d_ws, size_t ws_size,
                              hipStream_t stream)
{
    const float* state = (const float*)d_in[0];   // [512,128]
    const float* mem   = (const float*)d_in[1];   // [100000,128]

    int*   out_idx = (int*)d_out;                           // [512*100000] int32
    float* out_w   = (float*)d_out + (size_t)BATCH * MEMN;  // [512*100000] f32

    // workspace carve: inv norms + two ping-pong {key,idx} buffers
    char*  ws    = (char*)d_ws;
    float* inv_s = (float*)ws;                     // 512 floats
    float* inv_m = inv_s + BATCH;                  // 100000 floats
    size_t off   = (((size_t)(BATCH + MEMN) * sizeof(float)) + 255) & ~(size_t)255;
    uint2* bufA  = (uint2*)(ws + off);
    uint2* bufB  = bufA + (size_t)BATCH * MEMN;

    memonet_inv_norm_kernel<<<(BATCH + 7) / 8, 256, 0, stream>>>(state, inv_s, BATCH);
    memonet_inv_norm_kernel<<<(MEMN + 7) / 8, 256, 0, stream>>>(mem, inv_m, MEMN);

    // 512/16 = 32 tiles (M), 100000/16 = 6250 tiles (N); one wave32 per tile
    memonet_cosine_wmma_kernel<<<dim3(MEMN / 16, BATCH / 16), 32, 0, stream>>>(
        state, mem, inv_s, inv_m, out_w, bufA);

    const size_t ldsBytes = (size_t)(RADIX * HIST_STRIDE + RADIX) * sizeof(unsigned);
    memonet_radix_pass_kernel<<<BATCH, SORT_THREADS, ldsBytes, stream>>>(bufA, bufB, nullptr, 0);
    memonet_radix_pass_kernel<<<BATCH, SORT_THREADS, ldsBytes, stream>>>(bufB, bufA, nullptr, 8);
    memonet_radix_pass_kernel<<<BATCH, SORT_THREADS, ldsBytes, stream>>>(bufA, bufB, nullptr, 16);
    memonet_radix_pass_kernel<<<BATCH, SORT_THREADS, ldsBytes, stream>>>(bufB, nullptr, out_idx, 24);
}